// PositionalSampleEncoding_65944927863074
// MI455X (gfx1250) — compile-verified
//
#include <hip/hip_runtime.h>
#include <hip/hip_bf16.h>

#define NPTS 65536
#define MQ   1024
#define NS1  32
#define NS2  64
#define R1SQ 0.01f
#define R2SQ 0.04f

typedef __attribute__((ext_vector_type(2))) float v2f;
typedef __attribute__((ext_vector_type(8))) float v8f;

// ---- weight block layout inside LDS (704 floats per branch) ----
// wA[96] gA[16] bA[16] wB[512] gB[32] bB[32]
__device__ __forceinline__ void cp_lds(float* dst, const float* src, int n, int lane) {
    for (int t = lane; t < n; t += 32) dst[t] = src[t];
}

// Per-neighbor 6->16->32 MLP, weights in LDS (uniform-address broadcast reads).
__device__ __forceinline__ void mlp_point(const float* __restrict__ P, int i,
                                          float qx, float qy, float qz,
                                          const float* __restrict__ sw, float inv,
                                          float out32[32]) {
    const float gx = P[i * 3 + 0], gy = P[i * 3 + 1], gz = P[i * 3 + 2];
    float f[6] = {gx - qx, gy - qy, gz - qz, gx, gy, gz};
    const float* wA = sw;
    const float* gA = sw + 96;
    const float* bA = sw + 112;
    const float* wB = sw + 128;
    const float* gB = sw + 640;
    const float* bB = sw + 672;
    float ya[16];
#pragma unroll
    for (int o = 0; o < 16; ++o) {
        float acc = 0.f;
#pragma unroll
        for (int c = 0; c < 6; ++c) acc = fmaf(f[c], wA[o * 6 + c], acc);
        acc = fmaf(acc * inv, gA[o], bA[o]);
        ya[o] = fmaxf(acc, 0.f);
    }
#pragma unroll
    for (int o = 0; o < 32; ++o) {
        float acc = 0.f;
#pragma unroll
        for (int c = 0; c < 16; ++c) acc = fmaf(ya[c], wB[o * 16 + c], acc);
        acc = fmaf(acc * inv, gB[o], bB[o]);
        out32[o] = fmaxf(acc, 0.f);
    }
}

// One wave32 per query point: early-exit ordered ball query for both radii,
// fused gather + MLP + max-pool. Writes h row (64 floats) to workspace.
__global__ __launch_bounds__(32) void ball_mlp_kernel(
    const float* __restrict__ pts1,
    const float* __restrict__ w1a, const float* __restrict__ g1a, const float* __restrict__ b1a,
    const float* __restrict__ w1b, const float* __restrict__ g1b, const float* __restrict__ b1b,
    const float* __restrict__ w2a, const float* __restrict__ g2a, const float* __restrict__ b2a,
    const float* __restrict__ w2b, const float* __restrict__ g2b, const float* __restrict__ b2b,
    float* __restrict__ H) {
    __shared__ int   sIdx1[NS1];
    __shared__ int   sIdx2[NS2];
    __shared__ float sW[1408];        // two 704-float weight blocks
    __shared__ float sF[32][33];      // max-pool transpose buffer (pad 33)

    const int lane = threadIdx.x;
    const int r = blockIdx.x;         // global row: b*1024 + m
    const int b = r >> 10;
    const int m = r & 1023;

    // stage weights once
    cp_lds(sW + 0, w1a, 96, lane);
    cp_lds(sW + 96, g1a, 16, lane);
    cp_lds(sW + 112, b1a, 16, lane);
    cp_lds(sW + 128, w1b, 512, lane);
    cp_lds(sW + 640, g1b, 32, lane);
    cp_lds(sW + 672, b1b, 32, lane);
    cp_lds(sW + 704 + 0, w2a, 96, lane);
    cp_lds(sW + 704 + 96, g2a, 16, lane);
    cp_lds(sW + 704 + 112, b2a, 16, lane);
    cp_lds(sW + 704 + 128, w2b, 512, lane);
    cp_lds(sW + 704 + 640, g2b, 32, lane);
    cp_lds(sW + 704 + 672, b2b, 32, lane);

    // query point: flat = (4+8*(m/32))*256 + (4+8*(m%32))
    const int fi = (4 + 8 * (m >> 5)) * 256 + (4 + 8 * (m & 31));
    const float* P = pts1 + (size_t)b * NPTS * 3;
    const float qx = P[fi * 3 + 0], qy = P[fi * 3 + 1], qz = P[fi * 3 + 2];
    const float q2 = qx * qx + qy * qy + qz * qz;

    // ---- ordered early-exit ball query (both radii in one scan) ----
    int cnt1 = 0, cnt2 = 0;
    const unsigned lt = (1u << lane) - 1u;
    for (int base = 0; base < NPTS; base += 32) {
        const int i = base + lane;
        const float px = P[i * 3 + 0], py = P[i * 3 + 1], pz = P[i * 3 + 2];
        // mimic reference rounding: d2 = (q2 + p2) - 2*dot
        const float p2 = px * px + py * py + pz * pz;
        const float dot = px * qx + py * qy + pz * qz;
        const float d2 = (q2 + p2) - 2.0f * dot;
        const unsigned mask1 = (unsigned)__ballot(d2 < R1SQ);
        const unsigned mask2 = (unsigned)__ballot(d2 < R2SQ);
        if (cnt1 < NS1 && ((mask1 >> lane) & 1u)) {
            const int slot = cnt1 + __popc(mask1 & lt);
            if (slot < NS1) sIdx1[slot] = i;
        }
        if (cnt2 < NS2 && ((mask2 >> lane) & 1u)) {
            const int slot = cnt2 + __popc(mask2 & lt);
            if (slot < NS2) sIdx2[slot] = i;
        }
        cnt1 += __popc(mask1);
        cnt2 += __popc(mask2);
        if (cnt1 >= NS1 && cnt2 >= NS2) break;
    }
    __syncthreads();
    // fill trailing slots with first valid index (or 0 if none) — ref semantics
    const int c1 = min(cnt1, NS1), c2 = min(cnt2, NS2);
    const int fill1 = (c1 > 0) ? sIdx1[0] : 0;
    const int fill2 = (c2 > 0) ? sIdx2[0] : 0;
    if (lane >= c1) sIdx1[lane] = fill1;
    for (int s = lane; s < NS2; s += 32)
        if (s >= c2) sIdx2[s] = fill2;
    __syncthreads();

    const float inv = 1.0f / sqrtf(1.0f + 1e-5f);

    // ---- branch 1: S=32, one neighbor per lane ----
    {
        float v[32];
        mlp_point(P, sIdx1[lane], qx, qy, qz, sW, inv, v);
#pragma unroll
        for (int o = 0; o < 32; ++o) sF[lane][o] = v[o];
        __syncthreads();
        float mx = sF[0][lane];
#pragma unroll
        for (int s = 1; s < 32; ++s) mx = fmaxf(mx, sF[s][lane]);
        H[(size_t)r * 64 + lane] = mx;
        __syncthreads();
    }

    // ---- branch 2: S=64, two neighbors per lane ----
    {
        float va[32], vb[32];
        mlp_point(P, sIdx2[lane], qx, qy, qz, sW + 704, inv, va);
        mlp_point(P, sIdx2[lane + 32], qx, qy, qz, sW + 704, inv, vb);
#pragma unroll
        for (int o = 0; o < 32; ++o) sF[lane][o] = fmaxf(va[o], vb[o]);
        __syncthreads();
        float mx = sF[0][lane];
#pragma unroll
        for (int s = 1; s < 32; ++s) mx = fmaxf(mx, sF[s][lane]);
        H[(size_t)r * 64 + 32 + lane] = mx;
    }
}

// Head GEMM: out(2048x128) = H(2048x64) @ w3^T(64x128) + b3,
// exact f32 via V_WMMA_F32_16X16X4_F32; one wave per 16x16 tile.
__global__ __launch_bounds__(32) void head_wmma_kernel(
    const float* __restrict__ H, const float* __restrict__ w3,
    const float* __restrict__ b3, float* __restrict__ out) {
    const int mt = blockIdx.x;      // 0..127
    const int nt = blockIdx.y;      // 0..7
    const int lane = threadIdx.x;
    const int half = lane >> 4;     // 0: K,K+1 rows/cols 0-15 ; 1: K+2,K+3
    const int l16 = lane & 15;
    const int m0 = mt * 16, n0 = nt * 16;
    const float* Arow = H + (size_t)(m0 + l16) * 64;          // A[m][k]
    const float* Brow = w3 + (size_t)(n0 + l16) * 64;         // B[k][n] = w3[n*64+k]

    v8f acc = {};
#pragma unroll
    for (int k0 = 0; k0 < 64; k0 += 4) {
        const int k = k0 + 2 * half;
        v2f a, bb;
        a.x = Arow[k];
        a.y = Arow[k + 1];
        bb.x = Brow[k];
        bb.y = Brow[k + 1];
        acc = __builtin_amdgcn_wmma_f32_16x16x4_f32(
            /*neg_a=*/false, a, /*neg_b=*/false, bb,
            /*c_mod=*/(short)0, acc, /*reuse_a=*/false, /*reuse_b=*/false);
    }

    const float bias = b3[n0 + l16];
#pragma unroll
    for (int j = 0; j < 8; ++j) {
        const int mrow = m0 + j + 8 * half;   // C/D layout: VGPR j -> M=j (lanes0-15), M=8+j (lanes16-31)
        out[(size_t)mrow * 128 + n0 + l16] = acc[j] + bias;
    }
}

extern "C" void kernel_launch(void* const* d_in, const int* in_sizes, int n_in,
                              void* d_out, int out_size, void* d_ws, size_t ws_size,
                              hipStream_t stream) {
    const float* pts1 = (const float*)d_in[0];
    const float* w1a = (const float*)d_in[1];
    const float* g1a = (const float*)d_in[2];
    const float* b1a = (const float*)d_in[3];
    const float* w1b = (const float*)d_in[4];
    const float* g1b = (const float*)d_in[5];
    const float* b1b = (const float*)d_in[6];
    const float* w2a = (const float*)d_in[7];
    const float* g2a = (const float*)d_in[8];
    const float* b2a = (const float*)d_in[9];
    const float* w2b = (const float*)d_in[10];
    const float* g2b = (const float*)d_in[11];
    const float* b2b = (const float*)d_in[12];
    const float* w3 = (const float*)d_in[13];
    const float* b3 = (const float*)d_in[14];
    float* out = (float*)d_out;
    float* H = (float*)d_ws;  // 2048 x 64 floats = 512 KB scratch

    ball_mlp_kernel<<<dim3(2048), dim3(32), 0, stream>>>(
        pts1, w1a, g1a, b1a, w1b, g1b, b1b,
        w2a, g2a, b2a, w2b, g2b, b2b, H);

    head_wmma_kernel<<<dim3(128, 8), dim3(32), 0, stream>>>(H, w3, b3, out);
}